// GNNEncoder_1537598292354
// MI455X (gfx1250) — compile-verified
//
#include <hip/hip_runtime.h>

typedef __attribute__((ext_vector_type(2))) float v2f;
typedef __attribute__((ext_vector_type(8))) float v8f;

#define HD   128           // hidden dim (static for WMMA tiling)
#define LDA  132           // padded LDS row stride (floats) to kill bank conflicts

// ---------------------------------------------------------------- utilities
__global__ void zero_f32_kernel(float* __restrict__ p, long n) {
    long i = (long)blockIdx.x * blockDim.x + threadIdx.x;
    if (i < n) p[i] = 0.0f;
}

__global__ void degree_kernel(const int* __restrict__ src, const int* __restrict__ dst,
                              float* __restrict__ out_deg, float* __restrict__ in_deg, int E) {
    int e = blockIdx.x * blockDim.x + threadIdx.x;
    if (e < E) {
        atomicAdd(&out_deg[src[e]], 1.0f);
        atomicAdd(&in_deg[dst[e]], 1.0f);
    }
}

// deg -> deg^{-1/2} in place (deg clamped to >= 1)
__global__ void norm_kernel(float* __restrict__ out_nrm, float* __restrict__ in_nrm, int N) {
    int i = blockIdx.x * blockDim.x + threadIdx.x;
    if (i < N) {
        float od = out_nrm[i]; od = od < 1.0f ? 1.0f : od;
        float id = in_nrm[i];  id = id < 1.0f ? 1.0f : id;
        out_nrm[i] = rsqrtf(od);
        in_nrm[i]  = rsqrtf(id);
    }
}

// ------------------------------------------------- edge gather + scatter-add
// One wave per edge; lane l moves cols [4l, 4l+3] (32 lanes x float4 = 128).
__global__ __launch_bounds__(256)
void scatter_kernel(const float* __restrict__ h, const float* __restrict__ out_nrm,
                    const int* __restrict__ src, const int* __restrict__ dst,
                    float* __restrict__ agg, int E) {
    long gtid = (long)blockIdx.x * blockDim.x + threadIdx.x;
    int  e    = (int)(gtid >> 5);
    int  lane = (int)(gtid & 31);
    if (e < E) {
        int s = src[e];
        int d = dst[e];
        float sc = out_nrm[s];
        const float4* hp = (const float4*)(h + (size_t)s * HD);
        float4 v = hp[lane];
        float* ap = agg + (size_t)d * HD + lane * 4;
        atomicAdd(ap + 0, v.x * sc);
        atomicAdd(ap + 1, v.y * sc);
        atomicAdd(ap + 2, v.z * sc);
        atomicAdd(ap + 3, v.w * sc);
    }
}

// ------------------------------------------------------------- fused GEMM
// out = relu( (agg * in_norm[:,None]) @ W + bias ) + resid
// Block: 16 rows x 128 cols, 8 waves, each wave = one 16x16 tile via
// 32 chained V_WMMA_F32_16X16X4_F32 (exact f32 path).
__global__ __launch_bounds__(256)
void gemm_bias_relu_res_kernel(const float* __restrict__ agg,
                               const float* __restrict__ in_nrm,
                               const float* __restrict__ W,
                               const float* __restrict__ bias,
                               const float* __restrict__ resid,
                               float* __restrict__ out, int N) {
    __shared__ float As[16 * LDA];

    int rowbase = blockIdx.x * 16;
    int t = threadIdx.x;

    // Stage scaled A panel (16 x 128) into LDS: 2048 elements, 8 per thread.
    for (int i = t; i < 16 * HD; i += 256) {
        int r = i >> 7;       // i / 128
        int c = i & (HD - 1); // i % 128
        int row = rowbase + r;
        float v = 0.0f;
        if (row < N) v = agg[(size_t)row * HD + c] * in_nrm[row];
        As[r * LDA + c] = v;
    }
    __syncthreads();

    int wave = t >> 5;        // column tile 0..7
    int lane = t & 31;
    int m    = lane & 15;     // row (A) / col-within-tile (B,D)
    int hi   = lane >> 4;     // half-wave selector
    int n    = wave * 16 + m; // global output column

    v8f c = {};
    const float* Wb = W + n;  // column n of W (row-major [K][N])

    #pragma unroll
    for (int k = 0; k < HD / 4; ++k) {
        int kk = 4 * k + 2 * hi;
        v2f a, b;
        a.x = As[m * LDA + kk];
        a.y = As[m * LDA + kk + 1];
        b.x = Wb[(size_t)kk * HD];
        b.y = Wb[(size_t)(kk + 1) * HD];
        // D = A(16x4) * B(4x16) + C   (8 args: neg_a, A, neg_b, B, c_mod, C, reuse_a, reuse_b)
        c = __builtin_amdgcn_wmma_f32_16x16x4_f32(false, a, false, b, (short)0, c, false, false);
    }

    float bv = bias[n];
    #pragma unroll
    for (int j = 0; j < 8; ++j) {
        int row = rowbase + j + 8 * hi;   // D VGPR j -> row j (lo half) / j+8 (hi half)
        if (row < N) {
            float v = c[j] + bv;
            v = v > 0.0f ? v : 0.0f;
            out[(size_t)row * HD + n] = v + resid[(size_t)row * HD + n];
        }
    }
}

// -------------------------------------------------- nodes_src mask (B x n)
__global__ void mask_kernel(float* __restrict__ mask, const int* __restrict__ node_idx,
                            int N, int B) {
    int i = blockIdx.x * blockDim.x + threadIdx.x;
    if (i < N) {
        int n_per  = N / B;
        int b      = i / n_per;
        int local  = i - b * n_per;
        float v = 1.0f;
        if (b == B - 1) {
            int last = node_idx[B] - node_idx[B - 1];
            if (local >= last) v = 0.0f;
        }
        mask[i] = v;
    }
}

// ---------------------------------------------------------------- launcher
extern "C" void kernel_launch(void* const* d_in, const int* in_sizes, int n_in,
                              void* d_out, int out_size, void* d_ws, size_t ws_size,
                              hipStream_t stream) {
    const float* node_feats = (const float*)d_in[0];
    const int*   edge_src   = (const int*)  d_in[1];
    const int*   edge_dst   = (const int*)  d_in[2];
    const int*   node_idx   = (const int*)  d_in[3];
    const float* W1         = (const float*)d_in[5];
    const float* b1         = (const float*)d_in[6];
    const float* W2         = (const float*)d_in[7];
    const float* b2         = (const float*)d_in[8];

    const int Hdim = in_sizes[6];          // 128
    const int N    = in_sizes[0] / Hdim;   // 100000
    const int E    = in_sizes[1];          // 3200000
    const int B    = in_sizes[4];          // 8
    (void)Hdim; (void)n_in; (void)out_size; (void)ws_size;

    // workspace layout (floats): out_norm[N] | in_norm[N] | agg[N*H] | h1[N*H]
    float* ws       = (float*)d_ws;
    float* out_nrm  = ws;
    float* in_nrm   = ws + N;
    float* agg      = ws + 2 * (size_t)N;
    float* h1       = agg + (size_t)N * HD;
    float* h2       = (float*)d_out;                 // [N, H] final features
    float* mask     = h2 + (size_t)N * HD;           // [B, n] mask

    const long NH = (long)N * HD;

    // 1) zero deg arrays + agg in one pass
    {
        long zn = 2L * N + NH;
        zero_f32_kernel<<<(unsigned)((zn + 255) / 256), 256, 0, stream>>>(ws, zn);
    }
    // 2) degrees, 3) norms
    degree_kernel<<<(E + 255) / 256, 256, 0, stream>>>(edge_src, edge_dst, out_nrm, in_nrm, E);
    norm_kernel<<<(N + 255) / 256, 256, 0, stream>>>(out_nrm, in_nrm, N);

    const unsigned scatter_blocks = (unsigned)(((long)E * 32 + 255) / 256);
    const unsigned gemm_blocks    = (unsigned)((N + 15) / 16);

    // ---- layer 1: h1 = relu((A node_feats) @ W1 + b1) + node_feats
    scatter_kernel<<<scatter_blocks, 256, 0, stream>>>(node_feats, out_nrm,
                                                       edge_src, edge_dst, agg, E);
    gemm_bias_relu_res_kernel<<<gemm_blocks, 256, 0, stream>>>(agg, in_nrm, W1, b1,
                                                               node_feats, h1, N);

    // ---- layer 2: h2 = relu((A h1) @ W2 + b2) + h1
    zero_f32_kernel<<<(unsigned)((NH + 255) / 256), 256, 0, stream>>>(agg, NH);
    scatter_kernel<<<scatter_blocks, 256, 0, stream>>>(h1, out_nrm,
                                                       edge_src, edge_dst, agg, E);
    gemm_bias_relu_res_kernel<<<gemm_blocks, 256, 0, stream>>>(agg, in_nrm, W2, b2,
                                                               h1, h2, N);

    // ---- nodes_src mask
    mask_kernel<<<(N + 255) / 256, 256, 0, stream>>>(mask, node_idx, N, B);
}